// GaussianSpot_33277406609740
// MI455X (gfx1250) — compile-verified
//
#include <hip/hip_runtime.h>

// GaussianSpot rendering on MI455X (gfx1250, wave32).
//
// img[b,f,d1,d2] = sum_k amp_k * ex_k[d1] * ey_k[d2] + bg      (separable!)
// One V_WMMA_F32_16X16X4_F32 renders an entire 14x14 tile as a rank-2 outer
// product plus background:
//    A(16x4): col0 = amp0*ex0[i], col1 = amp1*ex1[i], col2 = 1, col3 = 0
//    B(4x16): row0 = ey0[j],      row1 = ey1[j],      row2 = bg, row3 = 0
//
// Store-bandwidth bound (205 MB out @ 23.3 TB/s ~ 9.5 us floor). This rev
// trims VALU to stay under that floor:
//  - v_rcp_f32 instead of IEEE-div chains (shared between i2w and amp)
//  - exp2 with log2e folded into the coefficient (3 ops per Gaussian factor)
//  - packed 14-wide LDS tile: readback addressing is pure immediate offsets
//  - readfirstlane-scalarized tile id: per-tile params load via s_load (SMEM)

typedef __attribute__((ext_vector_type(2))) float v2f;
typedef __attribute__((ext_vector_type(8))) float v8f;

namespace {
constexpr int DD   = 14;               // image side
constexpr int Fk   = 512;              // frames
constexpr int Bk   = 512;              // AOIs
constexpr int N_Ft = 1024;             // target_locs frame-table depth
constexpr int PIX  = DD * DD;          // 196
constexpr int WPB  = 8;                // waves per block (256 threads)
}

__global__ __launch_bounds__(256) void gspot_wmma_kernel(
    const float* __restrict__ height,   // [2,B,F]
    const float* __restrict__ width,    // [2,B,F]
    const float* __restrict__ xs,       // [2,B,F]
    const float* __restrict__ ys,       // [2,B,F]
    const float* __restrict__ bg,       // [B,F]
    const float* __restrict__ tlocs,    // [N_T,N_F,2]
    const int*   __restrict__ n_idx,    // [B,1]
    const int*   __restrict__ f_tab,    // [F]
    float*       __restrict__ out,      // [1,B,F,14,14]
    int total_tiles)
{
    __shared__ float lds[WPB * PIX];    // packed 14x14 tile per wave (6.1 KB)

    const int lane = threadIdx.x & 31;
    const int wave = threadIdx.x >> 5;
    // Wave-uniform tile id, forced into an SGPR so every per-tile parameter
    // load below becomes an s_load (SMEM) instead of 32-lane broadcast vmem.
    const int t = __builtin_amdgcn_readfirstlane(blockIdx.x * WPB + wave);
    if (t >= total_tiles) return;       // wave-uniform guard

    const int b  = t >> 9;              // t / 512
    const int fc = t & (Fk - 1);
    const int kstride = Bk * Fk;

    // ---- per-tile scalars (SMEM) ----
    const int   n    = n_idx[b];
    const int   fi   = f_tab[fc];
    const float locx = tlocs[(n * N_Ft + fi) * 2 + 0];
    const float locy = tlocs[(n * N_Ft + fi) * 2 + 1];

    const float h0 = height[t],          h1 = height[kstride + t];
    const float w0 = width[t],           w1 = width[kstride + t];
    const float sx0 = locx + xs[t],      sx1 = locx + xs[kstride + t];
    const float sy0 = locy + ys[t],      sy1 = locy + ys[kstride + t];
    const float bgv = bg[t];

    // rcp(w^2) shared: i2w-with-log2e fold AND amplitude.
    // exp(-r^2/(2w^2)) = exp2(r^2 * c),  c = -log2(e)/2 * rcp(w^2)
    const float r0 = __builtin_amdgcn_rcpf(w0 * w0);
    const float r1 = __builtin_amdgcn_rcpf(w1 * w1);
    const float c0 = r0 * -0.7213475204444817f;
    const float c1 = r1 * -0.7213475204444817f;
    const float amp0 = h0 * r0 * 0.15915494309189535f;   // h/(2*pi*w^2)
    const float amp1 = h1 * r1 * 0.15915494309189535f;

    // ---- per-lane separable Gaussian factors (pixel index = lane & 15) ----
    const float dpix = (float)(lane & 15);
    const float dx0 = dpix - sx0, dx1 = dpix - sx1;
    const float dy0 = dpix - sy0, dy1 = dpix - sy1;
    const float ex0 = __builtin_amdgcn_exp2f(dx0 * dx0 * c0);
    const float ex1 = __builtin_amdgcn_exp2f(dx1 * dx1 * c1);
    const float ey0 = __builtin_amdgcn_exp2f(dy0 * dy0 * c0);
    const float ey1 = __builtin_amdgcn_exp2f(dy1 * dy1 * c1);

    // ---- pack WMMA operands ----
    // A 16x4 (MxK): VGPR0 = K0 (lanes 0-15, M=lane) / K2 (lanes 16-31)
    //               VGPR1 = K1                      / K3
    // B 4x16 (KxN): VGPR0 = row K0 (lanes 0-15, N=lane) / row K2 (lanes 16-31)
    //               VGPR1 = row K1                      / row K3
    const bool lo = lane < 16;
    v2f amat, bmat;
    amat.x = lo ? amp0 * ex0 : 1.0f;    // K0 col | K2 col (=1 carries bg)
    amat.y = lo ? amp1 * ex1 : 0.0f;    // K1 col | K3 col
    bmat.x = lo ? ey0        : bgv;     // K0 row | K2 row (=bg)
    bmat.y = lo ? ey1        : 0.0f;    // K1 row | K3 row

    v8f cacc = {};
    v8f dtile = __builtin_amdgcn_wmma_f32_16x16x4_f32(
        /*neg_a=*/false, amat, /*neg_b=*/false, bmat,
        /*c_mod=*/(short)0, cacc, /*reuse_a=*/false, /*reuse_b=*/false);

    // ---- stage PACKED 14-wide tile to wave-private LDS ----
    // D layout: lanes 0-15 hold rows 0-7 (VGPR r = row r), lanes 16-31 rows 8-15.
    // Rows 0..5 of each half need only col<14; rows 6,7 of the hi half (image
    // rows 14,15) are dropped entirely.
    float* tile = &lds[wave * PIX];
    const int  col   = lane & 15;
    const int  wb    = (lo ? 0 : 8 * DD) + col;  // packed dword index of row rbase
    const bool okA   = col < DD;                  // rows rbase+0 .. rbase+5 (+6,7 lo)
    const bool okB   = okA && lo;                 // rows rbase+6, rbase+7
#pragma unroll
    for (int r = 0; r < 8; ++r) {
        const bool ok = (r < 6) ? okA : okB;
        if (ok) tile[wb + r * DD] = dtile[r];
    }
    // same-wave LDS ops are in-order; compiler inserts s_wait_dscnt for reads.

    // ---- coalesced writeback: 196 contiguous dwords, immediate offsets ----
    const long base = (long)t * PIX;
    const float* tr = &lds[wave * PIX];
#pragma unroll
    for (int it = 0; it < 7; ++it) {
        const int idx = it * 32 + lane;
        if (idx < PIX)                  // folds: always true for it<6, lane<4 at it=6
            out[base + idx] = tr[idx];
    }
}

extern "C" void kernel_launch(void* const* d_in, const int* in_sizes, int n_in,
                              void* d_out, int out_size, void* d_ws, size_t ws_size,
                              hipStream_t stream) {
    const float* height = (const float*)d_in[0];
    const float* width  = (const float*)d_in[1];
    const float* xs     = (const float*)d_in[2];
    const float* ys     = (const float*)d_in[3];
    const float* bg     = (const float*)d_in[4];
    const float* tlocs  = (const float*)d_in[5];
    const int*   n_idx  = (const int*)d_in[6];
    const int*   f_tab  = (const int*)d_in[7];
    float*       out    = (float*)d_out;

    const int total_tiles = in_sizes[4];               // B*F from background
    const int blocks = (total_tiles + WPB - 1) / WPB;

    gspot_wmma_kernel<<<blocks, 256, 0, stream>>>(
        height, width, xs, ys, bg, tlocs, n_idx, f_tab, out, total_tiles);
}